// SGPR_Geo_Baseline_58299886076590
// MI455X (gfx1250) — compile-verified
//
#include <hip/hip_runtime.h>
#include <math.h>

// ---------------- problem constants ----------------
#define NPTS 4096
#define NB   4
#define KNN  20
#define CIN  79          // 64 geo + 3 xyz + 12 sem
#define FDIM 128
#define BNEPS 1e-5f
#define LSLOPE 0.2f

typedef __attribute__((ext_vector_type(2))) float v2f;
typedef __attribute__((ext_vector_type(8))) float v8f;

// FP32 WMMA 16x16x4: A 16x4 (2 VGPR), B 4x16 (2 VGPR), C/D 16x16 f32 (8 VGPR)
__device__ __forceinline__ v8f wmma_f32x4(v2f a, v2f b, v8f c) {
  return __builtin_amdgcn_wmma_f32_16x16x4_f32(false, a, false, b, (short)0, c,
                                               false, false);
}

// ---------------- squared norms ----------------
template <int IC>
__global__ void sqnorm_kernel(const float* __restrict__ x, size_t bs,
                              float* __restrict__ xx) {
  int tid = blockIdx.x * blockDim.x + threadIdx.x;
  if (tid >= NB * NPTS) return;
  int b = tid / NPTS, n = tid % NPTS;
  const float* xb = x + (size_t)b * bs;
  float acc = 0.f;
#pragma unroll
  for (int c = 0; c < IC; ++c) {
    float v = xb[(size_t)c * NPTS + n];
    acc += v * v;
  }
  xx[tid] = acc;
}

// ---------------- fused distance GEMM + top-K ----------------
// One wave per (batch, 32-row tile): two A-fragment sets stay in VGPRs, each
// streamed 16-column fragment feeds TWO WMMAs (halves L2 re-read traffic).
// The 32x16 neg-distance tile goes to LDS; all 32 lanes maintain one sorted
// top-20 list each.
template <int IC>
__global__ __launch_bounds__(32) void knn_kernel(const float* __restrict__ x,
                                                 size_t bs,
                                                 const float* __restrict__ xx,
                                                 int* __restrict__ idxout) {
  constexpr int S = (IC + 3) / 4;
  const int lane = threadIdx.x;
  const int t = blockIdx.x;
  const int nt = t & 127;   // NPTS/32 = 128 row-tiles
  const int b = t >> 7;
  const int n0 = nt << 5;
  const int m = lane & 15;
  const int kb = (lane >> 4) << 1;   // K-pair base for this half-wave
  const float* xb = x + (size_t)b * bs;
  const float* xxb = xx + (size_t)b * NPTS;

  v2f afr[2][S];
#pragma unroll
  for (int h = 0; h < 2; ++h) {
#pragma unroll
    for (int s = 0; s < S; ++s) {
      int c = s * 4 + kb;
      int row = n0 + (h << 4) + m;
      afr[h][s][0] = (c < IC) ? xb[(size_t)c * NPTS + row] : 0.f;
      afr[h][s][1] = (c + 1 < IC) ? xb[(size_t)(c + 1) * NPTS + row] : 0.f;
    }
  }

  __shared__ float xxr[32];
  __shared__ float tile[32][16];
  __shared__ float topv[32][KNN];
  __shared__ int topi[32][KNN];
  xxr[lane] = xxb[n0 + lane];
  for (int k = 0; k < KNN; ++k) {
    topv[lane][k] = -3.4e38f;
    topi[lane][k] = 0;
  }
  __syncthreads();

  for (int ct = 0; ct < 256; ++ct) {
    const int nc = ct << 4;
    float xxc = xxb[nc + m];
    if (ct + 1 < 256)
      __builtin_prefetch(&xb[(size_t)kb * NPTS + nc + 16 + m], 0, 1);
    v8f acc0 = {}, acc1 = {};
#pragma unroll
    for (int s = 0; s < S; ++s) {
      int c = s * 4 + kb;
      v2f bf;
      bf[0] = (c < IC) ? xb[(size_t)c * NPTS + nc + m] : 0.f;
      bf[1] = (c + 1 < IC) ? xb[(size_t)(c + 1) * NPTS + nc + m] : 0.f;
      acc0 = wmma_f32x4(afr[0][s], bf, acc0);
      acc1 = wmma_f32x4(afr[1][s], bf, acc1);
    }
    const int mhi = (lane >> 4) << 3;
#pragma unroll
    for (int r = 0; r < 8; ++r) {
      int M = r + mhi;
      tile[M][m] = 2.f * acc0[r] - xxr[M] - xxc;        // rows n0..n0+15
      tile[16 + M][m] = 2.f * acc1[r] - xxr[16 + M] - xxc;  // rows +16..+31
    }
    __syncthreads();
    {
      const int row = lane;   // every lane owns one of the 32 rows
      for (int col = 0; col < 16; ++col) {
        float v = tile[row][col];
        if (v > topv[row][KNN - 1]) {
          int p = KNN - 1;
          while (p > 0 && topv[row][p - 1] < v) {
            topv[row][p] = topv[row][p - 1];
            topi[row][p] = topi[row][p - 1];
            --p;
          }
          topv[row][p] = v;
          topi[row][p] = nc + col;
        }
      }
    }
    __syncthreads();
  }
  for (int k = 0; k < KNN; ++k)
    idxout[((size_t)b * NPTS + n0 + lane) * KNN + k] = topi[lane][k];
}

// ---------------- fused gather + edge-conv GEMM + BN/leaky + max_k ----------
// One wave per (batch, 64-OC group, 16-point tile): each gathered edge-feature
// fragment (the expensive scattered loads) feeds FOUR WMMAs (4 OC tiles),
// cutting gather traffic 4x vs one-tile-per-wave. max-over-k stays in regs.
template <int IC, int OC>
__global__ __launch_bounds__(32) void edgeconv_kernel(
    const float* __restrict__ x, size_t bsIn, const int* __restrict__ idx,
    const float* __restrict__ W, const float* __restrict__ bias,
    const float* __restrict__ gamma, const float* __restrict__ beta,
    const float* __restrict__ mean, const float* __restrict__ var,
    float* __restrict__ out, size_t bsOut, int cOff) {
  constexpr int TWO = 2 * IC;
  constexpr int S = (TWO + 3) / 4;
  constexpr int NT = 4;               // 4 x 16 = 64 output channels per wave
  const int lane = threadIdx.x;
  int t = blockIdx.x;
  const int nt = t & 255;
  t >>= 8;
  const int og = t % (OC / 64);
  const int b = t / (OC / 64);
  const int n0 = nt << 4;
  const int ob = og << 6;             // base output channel of this block
  const int m = lane & 15;
  const int kb = (lane >> 4) << 1;
  const float* xb = x + (size_t)b * bsIn;

  __shared__ float s_sc[64], s_sh[64];
  for (int o = lane; o < 64; o += 32) {
    int oo = ob + o;
    float sc = gamma[oo] * rsqrtf(var[oo] + BNEPS);
    s_sc[o] = sc;
    s_sh[o] = (bias[oo] - mean[oo]) * sc + beta[oo];
  }
  __syncthreads();

  v8f amax[NT];
#pragma unroll
  for (int ot = 0; ot < NT; ++ot)
#pragma unroll
    for (int r = 0; r < 8; ++r) amax[ot][r] = -3.4e38f;

  for (int kk = 0; kk < KNN; ++kk) {
    const int nbr = idx[((size_t)b * NPTS + n0 + m) * KNN + kk];
    v8f acc[NT];
#pragma unroll
    for (int ot = 0; ot < NT; ++ot) acc[ot] = (v8f){};
#pragma unroll
    for (int s = 0; s < S; ++s) {
      int c0 = s * 4 + kb;
      v2f bf;                          // gathered edge feature, loaded ONCE
#pragma unroll
      for (int q = 0; q < 2; ++q) {
        int c = c0 + q;
        float v;
        if (c < IC)
          v = xb[(size_t)c * NPTS + nbr] - xb[(size_t)c * NPTS + n0 + m];
        else if (c < TWO)
          v = xb[(size_t)(c - IC) * NPTS + n0 + m];
        else
          v = 0.f;
        bf[q] = v;
      }
#pragma unroll
      for (int ot = 0; ot < NT; ++ot) {
        v2f a;
        int orow = ob + (ot << 4) + m;
        a[0] = (c0 < TWO) ? W[(size_t)orow * TWO + c0] : 0.f;
        a[1] = (c0 + 1 < TWO) ? W[(size_t)orow * TWO + c0 + 1] : 0.f;
        acc[ot] = wmma_f32x4(a, bf, acc[ot]);
      }
    }
    const int mhi = (lane >> 4) << 3;
#pragma unroll
    for (int ot = 0; ot < NT; ++ot) {
#pragma unroll
      for (int r = 0; r < 8; ++r) {
        int M = r + mhi;
        float y = acc[ot][r] * s_sc[(ot << 4) + M] + s_sh[(ot << 4) + M];
        y = (y >= 0.f) ? y : LSLOPE * y;
        amax[ot][r] = fmaxf(amax[ot][r], y);
      }
    }
  }
  const int mhi = (lane >> 4) << 3;
#pragma unroll
  for (int ot = 0; ot < NT; ++ot) {
#pragma unroll
    for (int r = 0; r < 8; ++r) {
      int M = r + mhi;
      out[(size_t)b * bsOut + (size_t)(cOff + ob + (ot << 4) + M) * NPTS + n0 + m] =
          amax[ot][r];
    }
  }
}

// ---------------- end conv: 128x384 GEMM + BN/leaky, transposed store -------
// One wave per (batch, 64-OC group, 16-point tile); each hcat fragment feeds
// four WMMAs.
__global__ __launch_bounds__(32) void endconv_kernel(
    const float* __restrict__ hc, const float* __restrict__ W,
    const float* __restrict__ gamma, const float* __restrict__ beta,
    const float* __restrict__ mean, const float* __restrict__ var,
    float* __restrict__ emb) {
  constexpr int CC = 3 * FDIM;   // 384
  constexpr int S = CC / 4;      // 96
  constexpr int NT = 4;
  const int lane = threadIdx.x;
  int t = blockIdx.x;
  const int nt = t & 255;
  t >>= 8;
  const int og = t & 1;
  const int b = t >> 1;
  const int n0 = nt << 4;
  const int ob = og << 6;
  const int m = lane & 15;
  const int kb = (lane >> 4) << 1;
  const float* hb = hc + (size_t)b * CC * NPTS;

  __shared__ float s_sc[64], s_sh[64];
  for (int o = lane; o < 64; o += 32) {
    int oo = ob + o;
    float sc = gamma[oo] * rsqrtf(var[oo] + BNEPS);
    s_sc[o] = sc;
    s_sh[o] = beta[oo] - mean[oo] * sc;
  }
  __syncthreads();

  v8f acc[NT];
#pragma unroll
  for (int ot = 0; ot < NT; ++ot) acc[ot] = (v8f){};
#pragma unroll
  for (int s = 0; s < S; ++s) {
    int c = s * 4 + kb;
    v2f bf;                      // hcat fragment, loaded once, used 4x
    bf[0] = hb[(size_t)c * NPTS + n0 + m];
    bf[1] = hb[(size_t)(c + 1) * NPTS + n0 + m];
#pragma unroll
    for (int ot = 0; ot < NT; ++ot) {
      v2f a;
      int orow = ob + (ot << 4) + m;
      a[0] = W[(size_t)orow * CC + c];
      a[1] = W[(size_t)orow * CC + c + 1];
      acc[ot] = wmma_f32x4(a, bf, acc[ot]);
    }
  }
  const int mhi = (lane >> 4) << 3;
#pragma unroll
  for (int ot = 0; ot < NT; ++ot) {
#pragma unroll
    for (int r = 0; r < 8; ++r) {
      int M = r + mhi;
      float y = acc[ot][r] * s_sc[(ot << 4) + M] + s_sh[(ot << 4) + M];
      y = (y >= 0.f) ? y : LSLOPE * y;
      emb[((size_t)b * NPTS + n0 + m) * FDIM + ob + (ot << 4) + M] = y;  // (B,N,F)
    }
  }
}

// ---------------- attention / head kernels ----------------
__global__ void mean_kernel(const float* __restrict__ emb, float* __restrict__ mo) {
  int tid = blockIdx.x * blockDim.x + threadIdx.x;
  if (tid >= NB * FDIM) return;
  int b = tid / FDIM, f = tid % FDIM;
  float acc = 0.f;
  for (int n = 0; n < NPTS; ++n) acc += emb[((size_t)b * NPTS + n) * FDIM + f];
  mo[tid] = acc * (1.f / NPTS);
}

__global__ void g_kernel(const float* __restrict__ m, const float* __restrict__ attW,
                         float* __restrict__ g) {
  int tid = blockIdx.x * blockDim.x + threadIdx.x;
  if (tid >= NB * FDIM) return;
  int b = tid / FDIM, fp = tid % FDIM;
  float acc = 0.f;
  for (int f = 0; f < FDIM; ++f) acc += m[b * FDIM + f] * attW[f * FDIM + fp];
  g[tid] = tanhf(acc);
}

__global__ void s_kernel(const float* __restrict__ emb, const float* __restrict__ g,
                         float* __restrict__ s) {
  int tid = blockIdx.x * blockDim.x + threadIdx.x;
  if (tid >= NB * NPTS) return;
  int b = tid >> 12, n = tid & (NPTS - 1);
  float acc = 0.f;
  for (int f = 0; f < FDIM; ++f)
    acc += emb[((size_t)b * NPTS + n) * FDIM + f] * g[b * FDIM + f];
  s[tid] = 1.f / (1.f + expf(-acc));
}

__global__ void pooled_kernel(const float* __restrict__ emb,
                              const float* __restrict__ s, float* __restrict__ p) {
  int tid = blockIdx.x * blockDim.x + threadIdx.x;
  if (tid >= NB * FDIM) return;
  int b = tid / FDIM, f = tid % FDIM;
  float acc = 0.f;
  for (int n = 0; n < NPTS; ++n)
    acc += emb[((size_t)b * NPTS + n) * FDIM + f] * s[b * NPTS + n];
  p[tid] = acc;
}

__global__ void head_kernel(const float* __restrict__ p1, const float* __restrict__ p2,
                            const float* __restrict__ tn_w, const float* __restrict__ tn_wb,
                            const float* __restrict__ tn_bias,
                            const float* __restrict__ fc1_w, const float* __restrict__ fc1_b,
                            const float* __restrict__ fc2_w, const float* __restrict__ fc2_b,
                            float* __restrict__ score) {
  __shared__ float sc[NB][16];
  int tid = threadIdx.x;
  if (tid < NB * 16) {
    int b = tid >> 4, t = tid & 15;
    float bil = 0.f;
    for (int f = 0; f < FDIM; ++f) {
      float df = p1[b * FDIM + f] - p2[b * FDIM + f];
      float acc = 0.f;
      for (int g = 0; g < FDIM; ++g) {
        float dg = p1[b * FDIM + g] - p2[b * FDIM + g];
        acc += tn_w[((size_t)f * FDIM + g) * 16 + t] * dg;
      }
      bil += df * acc;
    }
    float blk = 0.f;
    for (int j = 0; j < FDIM; ++j) blk += tn_wb[t * 256 + j] * p1[b * FDIM + j];
    for (int j = 0; j < FDIM; ++j) blk += tn_wb[t * 256 + FDIM + j] * p2[b * FDIM + j];
    float v = bil + blk + tn_bias[t];
    sc[b][t] = (v > 0.f) ? v : 0.f;
  }
  __syncthreads();
  if (tid < NB) {
    float acc2 = fc2_b[0];
    for (int i = 0; i < 16; ++i) {
      float h = fc1_b[i];
      for (int t2 = 0; t2 < 16; ++t2) h += fc1_w[i * 16 + t2] * sc[tid][t2];
      h = (h > 0.f) ? h : 0.f;
      acc2 += fc2_w[i] * h;
    }
    score[tid] = 1.f / (1.f + expf(-acc2));
  }
}

// ---------------- host-side branch driver ----------------
template <int IC0>
static void run_branch(const float* feats, int chOff, void* const* d_in, int pbase,
                       int hcatOff, float* xx, int* idx, float* xb0, float* xb1,
                       float* hcat, hipStream_t stream) {
  // sorted-key layer dict: 0=b 1=beta 2=gamma 3=mean 4=var 5=w
  auto P = [&](int l, int which) { return (const float*)d_in[pbase + l * 6 + which]; };
  const float* src = feats + (size_t)chOff * NPTS;
  const size_t bsF = (size_t)CIN * NPTS;
  const size_t bsX = (size_t)128 * NPTS;
  // layer 0: IC0 -> 64
  sqnorm_kernel<IC0><<<(NB * NPTS + 255) / 256, 256, 0, stream>>>(src, bsF, xx);
  knn_kernel<IC0><<<NB * 128, 32, 0, stream>>>(src, bsF, xx, idx);
  edgeconv_kernel<IC0, 64><<<NB * 1 * 256, 32, 0, stream>>>(
      src, bsF, idx, P(0, 5), P(0, 0), P(0, 2), P(0, 1), P(0, 3), P(0, 4), xb0, bsX, 0);
  // layer 1: 64 -> 64
  sqnorm_kernel<64><<<(NB * NPTS + 255) / 256, 256, 0, stream>>>(xb0, bsX, xx);
  knn_kernel<64><<<NB * 128, 32, 0, stream>>>(xb0, bsX, xx, idx);
  edgeconv_kernel<64, 64><<<NB * 1 * 256, 32, 0, stream>>>(
      xb0, bsX, idx, P(1, 5), P(1, 0), P(1, 2), P(1, 1), P(1, 3), P(1, 4), xb1, bsX, 0);
  // layer 2: 64 -> 128 (into concat buffer at channel offset)
  sqnorm_kernel<64><<<(NB * NPTS + 255) / 256, 256, 0, stream>>>(xb1, bsX, xx);
  knn_kernel<64><<<NB * 128, 32, 0, stream>>>(xb1, bsX, xx, idx);
  edgeconv_kernel<64, 128><<<NB * 2 * 256, 32, 0, stream>>>(
      xb1, bsX, idx, P(2, 5), P(2, 0), P(2, 2), P(2, 1), P(2, 3), P(2, 4), hcat,
      (size_t)384 * NPTS, hcatOff);
}

extern "C" void kernel_launch(void* const* d_in, const int* in_sizes, int n_in,
                              void* d_out, int out_size, void* d_ws, size_t ws_size,
                              hipStream_t stream) {
  (void)in_sizes; (void)n_in; (void)out_size; (void)ws_size;
  // d_in layout per JAX pytree (sorted-key) flattening of setup_inputs():
  // 0 features_1, 1 features_2, 2 att_w, 3-7 end{beta,gamma,mean,var,w},
  // 8 fc1_b, 9 fc1_w, 10 fc2_b, 11 fc2_w, 12-29 geo[3]{b,beta,gamma,mean,var,w},
  // 30-47 sem[...], 48 tn_bias, 49 tn_w, 50 tn_wb, 51-68 xyz[...]
  const float* f1 = (const float*)d_in[0];
  const float* f2 = (const float*)d_in[1];
  const float* attW = (const float*)d_in[2];
  const float* end_beta = (const float*)d_in[3];
  const float* end_gamma = (const float*)d_in[4];
  const float* end_mean = (const float*)d_in[5];
  const float* end_var = (const float*)d_in[6];
  const float* end_w = (const float*)d_in[7];
  const float* fc1_b = (const float*)d_in[8];
  const float* fc1_w = (const float*)d_in[9];
  const float* fc2_b = (const float*)d_in[10];
  const float* fc2_w = (const float*)d_in[11];
  const int GEO = 12, SEM = 30, TNB = 48, TNW = 49, TNWB = 50, XYZ = 51;

  float* ws = (float*)d_ws;
  float* xx = ws;            ws += NB * NPTS;
  int* idx = (int*)ws;       ws += NB * NPTS * KNN;
  float* xb0 = ws;           ws += (size_t)NB * 128 * NPTS;
  float* xb1 = ws;           ws += (size_t)NB * 128 * NPTS;
  float* hcat = ws;          ws += (size_t)NB * 384 * NPTS;
  float* emb1 = ws;          ws += (size_t)NB * NPTS * FDIM;
  float* emb2 = ws;          ws += (size_t)NB * NPTS * FDIM;
  float* mbuf = ws;          ws += NB * FDIM;
  float* gbuf = ws;          ws += NB * FDIM;
  float* pool1 = ws;         ws += NB * FDIM;
  float* pool2 = ws;         ws += NB * FDIM;

  float* score = (float*)d_out;
  float* s1 = score + NB;
  float* s2 = s1 + NB * NPTS;

  for (int pass = 0; pass < 2; ++pass) {
    const float* feats = pass ? f2 : f1;
    float* emb = pass ? emb2 : emb1;
    float* sout = pass ? s2 : s1;
    float* pooled = pass ? pool2 : pool1;

    run_branch<64>(feats, 0, d_in, GEO, 0, xx, idx, xb0, xb1, hcat, stream);
    run_branch<3>(feats, 64, d_in, XYZ, 128, xx, idx, xb0, xb1, hcat, stream);
    run_branch<12>(feats, 67, d_in, SEM, 256, xx, idx, xb0, xb1, hcat, stream);

    endconv_kernel<<<NB * 2 * 256, 32, 0, stream>>>(hcat, end_w, end_gamma,
                                                    end_beta, end_mean, end_var, emb);
    mean_kernel<<<2, 256, 0, stream>>>(emb, mbuf);
    g_kernel<<<2, 256, 0, stream>>>(mbuf, attW, gbuf);
    s_kernel<<<(NB * NPTS) / 256, 256, 0, stream>>>(emb, gbuf, sout);
    pooled_kernel<<<2, 256, 0, stream>>>(emb, sout, pooled);
  }

  head_kernel<<<1, 64, 0, stream>>>(pool1, pool2, (const float*)d_in[TNW],
                                    (const float*)d_in[TNWB], (const float*)d_in[TNB],
                                    fc1_w, fc1_b, fc2_w, fc2_b, score);
}